// MultigridSolver_70987219468525
// MI455X (gfx1250) — compile-verified
//
#include <hip/hip_runtime.h>
#include <math.h>

namespace {

constexpr float LAM   = 1.0f;
constexpr float OMEGA = 0.9f;
constexpr int NLEV = 4;

// Jacobi tiling: 256 threads (32x8, 8 waves of 32) compute a 32x32 tile,
// 4 points per thread. Halo tile is 34x34 (pitch 36 for bank spread).
constexpr int TILE   = 32;
constexpr int BY     = 8;
constexpr int PTS    = TILE / BY;    // 4 rows per thread
constexpr int TDIM   = TILE + 2;     // 34
constexpr int TPITCH = 36;

__global__ __launch_bounds__(256)
void k_zero(float* __restrict__ p, int count) {
    int i = blockIdx.x * 256 + threadIdx.x;
    if (i < count) p[i] = 0.0f;
}

__global__ __launch_bounds__(256)
void k_inv_diag(const float* __restrict__ mask, const float* __restrict__ cx,
                const float* __restrict__ cy, float* __restrict__ inv,
                int n, float inv_h2) {
    int gx = blockIdx.x * blockDim.x + threadIdx.x;
    int gy = blockIdx.y * blockDim.y + threadIdx.y;
    if (gx >= n || gy >= n) return;
    size_t id = (size_t)gy * n + gx;
    float cxe = cx[id];
    float cxw = (gx > 0) ? cx[id - 1] : 0.0f;
    float cyn = cy[id];
    float cys = (gy > 0) ? cy[id - n] : 0.0f;
    float diag = -((cxe + cxw) * inv_h2 + (cyn + cys) * inv_h2) - LAM;
    inv[id] = ((diag != 0.0f) ? (1.0f / diag) : 0.0f) * mask[id];
}

// One weighted-Jacobi sweep. u and mask 34x34 halo tiles are staged through
// LDS via the CDNA5 async global->LDS data mover (ASYNCcnt-tracked); the
// west coefficient cx_w comes from the neighbor lane via wave32 shuffle.
__global__ __launch_bounds__(256)
void k_jacobi(const float* __restrict__ u_in, float* __restrict__ u_out,
              const float* __restrict__ rhs, const float* __restrict__ mask,
              const float* __restrict__ cx, const float* __restrict__ cy,
              const float* __restrict__ inv_diag, int n, float inv_h2) {
    __shared__ float s_u[TDIM][TPITCH];
    __shared__ float s_m[TDIM][TPITCH];
    const int tx = threadIdx.x, ty = threadIdx.y;
    const int gx0 = blockIdx.x * TILE, gy0 = blockIdx.y * TILE;
    const int t = ty * TILE + tx;
    for (int idx = t; idx < TDIM * TDIM; idx += TILE * BY) {
        const int ly = idx / TDIM, lx = idx - ly * TDIM;
        const int gy = gy0 + ly - 1, gx = gx0 + lx - 1;
        if (gx >= 0 && gx < n && gy >= 0 && gy < n) {
            const size_t gid = (size_t)gy * n + gx;
            unsigned lu = (unsigned)(size_t)&s_u[ly][lx];
            unsigned lm = (unsigned)(size_t)&s_m[ly][lx];
            // CDNA5 async copy: global -> LDS, no VGPR round-trip
            asm volatile("global_load_async_to_lds_b32 %0, %1, off"
                         :: "v"(lu), "v"(u_in + gid) : "memory");
            asm volatile("global_load_async_to_lds_b32 %0, %1, off"
                         :: "v"(lm), "v"(mask + gid) : "memory");
        } else {
            s_u[ly][lx] = 0.0f;   // zero-padded ghost cells
            s_m[ly][lx] = 0.0f;
        }
    }
    asm volatile("s_wait_asynccnt 0x0" ::: "memory");
    __syncthreads();

    const int gx = gx0 + tx;
    const int li = tx + 1;
#pragma unroll
    for (int k = 0; k < PTS; ++k) {
        const int row = ty + k * BY;          // 0..31
        const int gy  = gy0 + row;
        const int lj  = row + 1;
        const size_t id = (size_t)gy * n + gx;
        const float m  = s_m[lj][li];
        const float ur = s_u[lj][li];         // raw (unmasked) u
        const float uc = ur * m;
        const float ue = s_u[lj][li + 1] * s_m[lj][li + 1];
        const float uw = s_u[lj][li - 1] * s_m[lj][li - 1];
        const float un = s_u[lj + 1][li] * s_m[lj + 1][li];
        const float us = s_u[lj - 1][li] * s_m[lj - 1][li];
        const float cxe = cx[id];
        // west coefficient = east coefficient of the lane to the left
        float cxw = __shfl_up(cxe, 1);
        if (tx == 0) cxw = (gx > 0) ? cx[id - 1] : 0.0f;
        const float cyn = cy[id];
        const float cys = (gy > 0) ? cy[id - n] : 0.0f;
        const float lap = (cxe * (ue - uc) - cxw * (uc - uw)) * inv_h2
                        + (cyn * (un - uc) - cys * (uc - us)) * inv_h2;
        const float Au = (lap - LAM * uc) * m;
        const float r  = rhs[id] - Au;
        u_out[id] = (ur + OMEGA * inv_diag[id] * r) * m;
    }
}

__global__ __launch_bounds__(256)
void k_residual(const float* __restrict__ u, const float* __restrict__ rhs,
                const float* __restrict__ mask, const float* __restrict__ cx,
                const float* __restrict__ cy, float* __restrict__ res,
                int n, float inv_h2) {
    int gx = blockIdx.x * blockDim.x + threadIdx.x;
    int gy = blockIdx.y * blockDim.y + threadIdx.y;
    if (gx >= n || gy >= n) return;
    size_t id = (size_t)gy * n + gx;
    float m  = mask[id];
    float uc = u[id] * m;
    float ue = (gx + 1 < n) ? u[id + 1] * mask[id + 1] : 0.0f;
    float uw = (gx > 0)     ? u[id - 1] * mask[id - 1] : 0.0f;
    float un = (gy + 1 < n) ? u[id + n] * mask[id + n] : 0.0f;
    float us = (gy > 0)     ? u[id - n] * mask[id - n] : 0.0f;
    float cxe = cx[id];
    float cxw = (gx > 0) ? cx[id - 1] : 0.0f;
    float cyn = cy[id];
    float cys = (gy > 0) ? cy[id - n] : 0.0f;
    float lap = (cxe * (ue - uc) - cxw * (uc - uw)) * inv_h2
              + (cyn * (un - uc) - cys * (uc - us)) * inv_h2;
    float Au = (lap - LAM * uc) * m;
    res[id] = (rhs[id] - Au) * m;
}

__global__ __launch_bounds__(256)
void k_restrict(const float* __restrict__ vf, const float* __restrict__ mf,
                const float* __restrict__ mc, float* __restrict__ rc, int nc) {
    int ic = blockIdx.x * blockDim.x + threadIdx.x;
    int jc = blockIdx.y * blockDim.y + threadIdx.y;
    if (ic >= nc || jc >= nc) return;
    int nf = nc * 2;
    size_t i00 = (size_t)(2 * jc) * nf + 2 * ic;
    float m00 = mf[i00],      m01 = mf[i00 + 1];
    float m10 = mf[i00 + nf], m11 = mf[i00 + nf + 1];
    float s = vf[i00] * m00 + vf[i00 + 1] * m01
            + vf[i00 + nf] * m10 + vf[i00 + nf + 1] * m11;
    float cnt = m00 + m01 + m10 + m11;
    size_t idc = (size_t)jc * nc + ic;
    rc[idc] = s / fmaxf(cnt, 1.0f) * mc[idc];
}

__global__ __launch_bounds__(256)
void k_prolong_add(const float* __restrict__ ec, const float* __restrict__ mc,
                   const float* __restrict__ mf, float* __restrict__ uf, int nf) {
    int gx = blockIdx.x * blockDim.x + threadIdx.x;
    int gy = blockIdx.y * blockDim.y + threadIdx.y;
    if (gx >= nf || gy >= nf) return;
    int nc = nf >> 1;
    int ic = gx >> 1, jc = gy >> 1;
    int si = (gx & 1) ? 1 : -1;
    int sj = (gy & 1) ? 1 : -1;
    auto fetch = [&](int j, int i, float& v, float& m) {
        if (i >= 0 && i < nc && j >= 0 && j < nc) {
            size_t id = (size_t)j * nc + i;
            m = mc[id];
            v = ec[id] * m;
        } else { v = 0.0f; m = 0.0f; }
    };
    float vc, mcv, vsi, msi, vsj, msj, vd, md;
    fetch(jc,      ic,      vc,  mcv);
    fetch(jc,      ic + si, vsi, msi);
    fetch(jc + sj, ic,      vsj, msj);
    fetch(jc + sj, ic + si, vd,  md);
    float w = 9.0f * vc  + 3.0f * vsi + 3.0f * vsj + vd;
    float d = 9.0f * mcv + 3.0f * msi + 3.0f * msj + md;
    size_t idf = (size_t)gy * nf + gx;
    uf[idf] += (w / fmaxf(d, 1.0f)) * mf[idf];
}

struct Level {
    const float* mask; const float* cx; const float* cy;
    float* inv; float* tmp; float* u; float* r;
    int n; float inv_h2;
};

static inline void smooth(const Level& L, int iters, hipStream_t s) {
    dim3 blk(TILE, BY), grd(L.n / TILE, L.n / TILE);
    for (int it = 0; it < iters; it += 2) {  // iters always even (2, 2, 22)
        k_jacobi<<<grd, blk, 0, s>>>(L.u, L.tmp, L.r, L.mask, L.cx, L.cy,
                                     L.inv, L.n, L.inv_h2);
        k_jacobi<<<grd, blk, 0, s>>>(L.tmp, L.u, L.r, L.mask, L.cx, L.cy,
                                     L.inv, L.n, L.inv_h2);
    }
}

} // namespace

extern "C" void kernel_launch(void* const* d_in, const int* in_sizes, int n_in,
                              void* d_out, int out_size, void* d_ws, size_t ws_size,
                              hipStream_t stream) {
    (void)n_in; (void)out_size; (void)ws_size;
    // fine grid size from mask0 element count (2048)
    int n0 = 1;
    while ((long long)n0 * n0 < (long long)in_sizes[0]) n0 <<= 1;

    // input order (setup_inputs dict): mask0,cx0,cy0, ..., mask3,cx3,cy3, rhs
    const float* rhs0 = (const float*)d_in[3 * NLEV];

    Level L[NLEV];
    float* ws = (float*)d_ws;
    size_t off = 0;
    for (int l = 0; l < NLEV; ++l) {
        int n = n0 >> l;
        L[l].mask = (const float*)d_in[3 * l + 0];
        L[l].cx   = (const float*)d_in[3 * l + 1];
        L[l].cy   = (const float*)d_in[3 * l + 2];
        L[l].n = n;
        L[l].inv_h2 = 1.0f / (float)(1 << (2 * l));    // 1/(2^l)^2
        L[l].inv = ws + off; off += (size_t)n * n;
        L[l].tmp = ws + off; off += (size_t)n * n;
        if (l == 0) {
            L[l].u = (float*)d_out;
            L[l].r = (float*)rhs0;   // never written at level 0
        } else {
            L[l].u = ws + off; off += (size_t)n * n;
            L[l].r = ws + off; off += (size_t)n * n;
        }
    }

    // inverse diagonals (every call: d_ws is not preserved)
    for (int l = 0; l < NLEV; ++l) {
        int n = L[l].n;
        dim3 blk(16, 16), grd(n / 16, n / 16);
        k_inv_diag<<<grd, blk, 0, stream>>>(L[l].mask, L[l].cx, L[l].cy,
                                            L[l].inv, n, L[l].inv_h2);
    }

    // u0 = 0
    {
        int cnt = n0 * n0;
        k_zero<<<(cnt + 255) / 256, 256, 0, stream>>>(L[0].u, cnt);
    }

    for (int cyc = 0; cyc < 2; ++cyc) {            // N_CYCLES
        // downward pass
        for (int l = 0; l < NLEV - 1; ++l) {
            smooth(L[l], 2, stream);               // N_PRE
            int n = L[l].n;
            dim3 blk(16, 16), grd(n / 16, n / 16);
            k_residual<<<grd, blk, 0, stream>>>(L[l].u, L[l].r, L[l].mask,
                                                L[l].cx, L[l].cy, L[l].tmp,
                                                n, L[l].inv_h2);
            int nc = L[l + 1].n;
            dim3 grdc(nc / 16, nc / 16);
            k_restrict<<<grdc, blk, 0, stream>>>(L[l].tmp, L[l].mask,
                                                 L[l + 1].mask, L[l + 1].r, nc);
            int cnt = nc * nc;
            k_zero<<<(cnt + 255) / 256, 256, 0, stream>>>(L[l + 1].u, cnt);
        }
        // coarsest: N_PRE + N_COARSE consecutive identical Jacobi sweeps
        smooth(L[NLEV - 1], 2 + 20, stream);
        // upward pass
        for (int l = NLEV - 2; l >= 0; --l) {
            int n = L[l].n;
            dim3 blk(16, 16), grd(n / 16, n / 16);
            k_prolong_add<<<grd, blk, 0, stream>>>(L[l + 1].u, L[l + 1].mask,
                                                   L[l].mask, L[l].u, n);
            smooth(L[l], 2, stream);               // N_POST
        }
    }
}